// VectorQuantizer_6279242187323
// MI455X (gfx1250) — compile-verified
//
#include <hip/hip_runtime.h>
#include <hip/hip_bf16.h>

// ---------- problem constants ----------
#define BDIM   64          // batch
#define EDIM   64          // embedding dim
#define HW     1024        // 32*32
#define NROWS  65536       // BDIM*HW flattened queries
#define NE     1024        // codebook entries
#define ZQ_SIZE   4194304  // 64*64*32*32
#define ENC_SIZE  67108864 // NROWS*NE

typedef float v2f __attribute__((ext_vector_type(2)));
typedef float v8f __attribute__((ext_vector_type(8)));

// -------------------------------------------------------------------------
// Kernel 1: fused distance matmul (WMMA f32 16x16x4) + argmin.
// One wave per 16-row tile of z. d(n,j) = ||z||^2 + ||e_j||^2 - 2 z.e_j ;
// ||z||^2 is row-constant so argmin uses (||e_j||^2 - 2*dot).
// -------------------------------------------------------------------------
__global__ __launch_bounds__(256)
void vq_argmin_kernel(const float* __restrict__ z,
                      const float* __restrict__ cb,
                      int* __restrict__ idx_out) {
    const int lane = threadIdx.x & 31;
    const int wave = threadIdx.x >> 5;
    const int tile = blockIdx.x * 8 + wave;   // 4096 tiles of 16 rows
    const int row0 = tile * 16;

    const int m = lane & 15;    // matrix row (A) / column (B,D) slot
    const int h = lane >> 4;    // half-wave selects K pair: +0 or +2

    // This lane's z row: n = row0 + m. NCHW layout -> dim stride is HW.
    const int n  = row0 + m;
    const int b  = n >> 10;
    const int hw = n & 1023;
    const float* zb = z + (size_t)b * EDIM * HW + hw;

    // Load A fragments once (reused across all 64 code tiles).
    // 32-bit A 16x4 layout: lanes 0-15 hold K = {4k,4k+1}, lanes 16-31 K = {4k+2,4k+3}.
    v2f a[16];
#pragma unroll
    for (int k = 0; k < 16; ++k) {
        const int c0 = 4 * k + 2 * h;
        a[k].x = zb[(size_t)c0 * HW];
        a[k].y = zb[(size_t)(c0 + 1) * HW];
    }

    float minv[8];
    int   mini[8];
#pragma unroll
    for (int r = 0; r < 8; ++r) { minv[r] = 3.4e38f; mini[r] = 0; }

    for (int t = 0; t < 64; ++t) {
        // B fragment source: codebook row (16t + m), dims {4k+2h, 4k+2h+1}
        // -> contiguous float2 in the row-major codebook.
        const float* cbrow = cb + (size_t)(t * 16 + m) * EDIM + 2 * h;

        v8f acc = {0.f, 0.f, 0.f, 0.f, 0.f, 0.f, 0.f, 0.f};
        float en_half = 0.f;   // this lane holds half the dims of code (16t+m)
#pragma unroll
        for (int k = 0; k < 16; ++k) {
            v2f bv = *(const v2f*)(cbrow + 4 * k);
            en_half += bv.x * bv.x + bv.y * bv.y;
            // D = A(16x4) * B(4x16) + C, f32 accumulate
            acc = __builtin_amdgcn_wmma_f32_16x16x4_f32(
                false, a[k], false, bv, (short)0, acc, false, false);
        }
        // complete ||e_j||^2: other half-wave has the remaining dims
        const float en = en_half + __shfl_xor(en_half, 16, 32);

        const int j = t * 16 + m;   // this lane's code column in D
#pragma unroll
        for (int r = 0; r < 8; ++r) {
            const float v = fmaf(-2.f, acc[r], en);
            if (v < minv[r]) { minv[r] = v; mini[r] = j; }   // first-hit wins ties
        }
    }

    // Cross-lane min-reduce within each 16-lane group (rows 0-7 in lanes 0-15,
    // rows 8-15 in lanes 16-31). Tie-break on lower index (matches argmin).
#pragma unroll
    for (int r = 0; r < 8; ++r) {
        float v = minv[r]; int i = mini[r];
#pragma unroll
        for (int mask = 1; mask < 16; mask <<= 1) {
            const float ov = __shfl_xor(v, mask, 32);
            const int   oi = __shfl_xor(i, mask, 32);
            if (ov < v || (ov == v && oi < i)) { v = ov; i = oi; }
        }
        minv[r] = v; mini[r] = i;
    }

    if (m == 0) {   // lanes 0 and 16 write rows [row0..row0+7] / [row0+8..row0+15]
        const int rowbase = row0 + 8 * h;
#pragma unroll
        for (int r = 0; r < 8; ++r) idx_out[rowbase + r] = mini[r];
    }
}

// -------------------------------------------------------------------------
// Kernel 2: gather z_q (NCHW), one-hot scatter, index-as-float, and per-block
// partial sum of (z - z_q)^2.  One thread per row.
// -------------------------------------------------------------------------
__global__ __launch_bounds__(256)
void vq_gather_loss_kernel(const float* __restrict__ z,
                           const float* __restrict__ cb,
                           const int* __restrict__ idx_in,
                           float* __restrict__ zq,
                           float* __restrict__ enc,
                           float* __restrict__ idxf,
                           float* __restrict__ partial) {
    const int n  = blockIdx.x * 256 + threadIdx.x;
    const int b  = n >> 10;
    const int hw = n & 1023;
    const int idx = idx_in[n];

    const float* crow = cb + (size_t)idx * EDIM;
    const float* zrow = z  + (size_t)b * EDIM * HW + hw;
    float*       qrow = zq + (size_t)b * EDIM * HW + hw;

    float s = 0.f;
#pragma unroll
    for (int c = 0; c < EDIM; ++c) {
        const float q  = crow[c];
        const float zv = zrow[(size_t)c * HW];
        const float d  = zv - q;
        s += d * d;
        qrow[(size_t)c * HW] = q;   // STE forward value == quantized value
    }

    idxf[n] = (float)idx;
    enc[(size_t)n * NE + idx] = 1.0f;   // region pre-zeroed by memset

    __shared__ float sm[256];
    sm[threadIdx.x] = s;
    __syncthreads();
    for (int st = 128; st > 0; st >>= 1) {
        if (threadIdx.x < st) sm[threadIdx.x] += sm[threadIdx.x + st];
        __syncthreads();
    }
    if (threadIdx.x == 0) partial[blockIdx.x] = sm[0];
}

// -------------------------------------------------------------------------
// Kernel 3: deterministic final reduction of 256 block partials.
// loss = (1 + beta) * mean((z - z_q)^2), beta = 0.25
// -------------------------------------------------------------------------
__global__ __launch_bounds__(256)
void vq_loss_final_kernel(const float* __restrict__ partial,
                          float* __restrict__ loss) {
    __shared__ float sm[256];
    sm[threadIdx.x] = partial[threadIdx.x];
    __syncthreads();
    for (int st = 128; st > 0; st >>= 1) {
        if (threadIdx.x < st) sm[threadIdx.x] += sm[threadIdx.x + st];
        __syncthreads();
    }
    if (threadIdx.x == 0) loss[0] = 1.25f * sm[0] / (float)ZQ_SIZE;
}

// -------------------------------------------------------------------------
extern "C" void kernel_launch(void* const* d_in, const int* in_sizes, int n_in,
                              void* d_out, int out_size, void* d_ws, size_t ws_size,
                              hipStream_t stream) {
    const float* z  = (const float*)d_in[0];   // [64,64,32,32] f32
    const float* cb = (const float*)d_in[1];   // [1024,64]     f32

    float* out  = (float*)d_out;
    float* loss = out;                                  // [1]
    float* zq   = out + 1;                              // [4194304]
    float* enc  = out + 1 + ZQ_SIZE;                    // [67108864]
    float* idxf = out + 1 + ZQ_SIZE + ENC_SIZE;         // [65536]

    int*   idx_ws  = (int*)d_ws;                                    // 65536 ints
    float* partial = (float*)((char*)d_ws + (size_t)NROWS * 4);     // 256 floats

    // Zero only the one-hot region (z_q / indices / loss are fully overwritten).
    hipMemsetAsync(enc, 0, (size_t)ENC_SIZE * sizeof(float), stream);

    // 4096 row-tiles, 8 waves (1 tile each) per 256-thread block.
    vq_argmin_kernel<<<512, 256, 0, stream>>>(z, cb, idx_ws);

    // 65536 rows, 1 thread each.
    vq_gather_loss_kernel<<<256, 256, 0, stream>>>(z, cb, idx_ws, zq, enc, idxf, partial);

    vq_loss_final_kernel<<<1, 256, 0, stream>>>(partial, loss);
}